// Encoder_5858335392034
// MI455X (gfx1250) — compile-verified
//
#include <hip/hip_runtime.h>
#include <hip/hip_bf16.h>

// ---------------------------------------------------------------------------
// VGAE encoder (2-layer GCN) for MI455X / gfx1250.
//  - GEMMs: V_WMMA_F32_16X16X4_F32 (fp32 = bit-faithful to the reference).
//    5 M-tiles (80 rows) per block (3125 tiles = 5^5 -> 625 blocks, no edge
//    cases, EXEC all-ones for every WMMA). Cuts W L2 traffic 5x vs 1 tile.
//  - A-strip staged into LDS in K-chunks of 128 with double buffering via
//    GLOBAL_LOAD_ASYNC_TO_LDS_B128 (ASYNCcnt + s_wait_asynccnt), falling
//    back to a synchronous float4 path if the builtin is absent.
//  - Edge aggregation: 1 thread per (edge, channel) fp32 atomics; h/agg
//    buffers (25.6 MB) are L2-resident (192 MB L2), atomics resolve on-chip.
// ---------------------------------------------------------------------------

#ifndef __has_builtin
#define __has_builtin(x) 0
#endif

typedef __attribute__((ext_vector_type(2))) float v2f;
typedef __attribute__((ext_vector_type(8))) float v8f;
typedef __attribute__((ext_vector_type(4))) int   v4i;

#if __has_builtin(__builtin_amdgcn_global_load_async_to_lds_b128)
#define HAVE_ASYNC_LDS 1
typedef __attribute__((address_space(1))) v4i v4i_g;   // global (addrspace 1)
typedef __attribute__((address_space(3))) v4i v4i_l;   // LDS    (addrspace 3)
#else
#define HAVE_ASYNC_LDS 0
#endif

__device__ __forceinline__ void async_wait0() {
#if HAVE_ASYNC_LDS
#if __has_builtin(__builtin_amdgcn_s_wait_asynccnt)
    __builtin_amdgcn_s_wait_asynccnt(0);
#else
    asm volatile("s_wait_asynccnt 0x0" ::: "memory");
#endif
#endif
}

// ---------------- elementwise helpers ----------------

__global__ void fill_f32_kernel(float* __restrict__ p, long long n, float v) {
    long long i = (long long)blockIdx.x * blockDim.x + threadIdx.x;
    if (i < n) p[i] = v;
}

__global__ void deg_edges_kernel(const int* __restrict__ dst,
                                 float* __restrict__ deg, int E) {
    int e = blockIdx.x * blockDim.x + threadIdx.x;
    if (e < E) atomicAdd(&deg[dst[e]], 1.0f);
}

__global__ void rsqrt_kernel(float* __restrict__ d, int n) {
    int i = blockIdx.x * blockDim.x + threadIdx.x;
    if (i < n) d[i] = rsqrtf(d[i]);   // deg >= 1 always (self loop)
}

// out[i] = (out[i] + h[i]*dis[row]^2 + bias[c]) , optional ReLU.
// Folds the self-loop contribution + bias + activation in one pass.
__global__ void finalize_kernel(float* __restrict__ out,
                                const float* __restrict__ h,
                                const float* __restrict__ dis,
                                const float* __restrict__ bias,
                                long long total, int C, int relu) {
    long long i = (long long)blockIdx.x * blockDim.x + threadIdx.x;
    if (i >= total) return;
    int c = (int)(i % C);
    long long row = i / C;
    float d = dis[row];
    float v = out[i] + h[i] * d * d + bias[c];
    if (relu) v = fmaxf(v, 0.0f);
    out[i] = v;
}

// ---------------- edge scatter:  out[dst,:] += h[src,:] * norm -------------

template <int C>
__global__ __launch_bounds__(256)
void scatter_kernel(const int* __restrict__ src, const int* __restrict__ dst,
                    const float* __restrict__ dis, const float* __restrict__ h,
                    float* __restrict__ out, int E) {
    constexpr int EPB = 256 / C;                 // edges per block
    const int c = threadIdx.x % C;
    const int e = blockIdx.x * EPB + threadIdx.x / C;  // wave-uniform
    if (e >= E) return;
    const int s = src[e];
    const int d = dst[e];
    const float nrm = dis[s] * dis[d];
    atomicAdd(&out[(size_t)d * C + c], h[(size_t)s * C + c] * nrm);
}

// ---------------- WMMA fp32 GEMM:  Out[M,NC] = A[M,K] @ W[NC,K]^T ----------
// Block: MT m-tiles (16*MT rows) x NC cols. NC/16 waves; wave w owns output
// cols [16w, 16w+16) across all MT row-tiles (reuses each B fragment MT x).
// A staged in LDS in K-chunks of KC, double-buffered (async when available).
// Requires M % (16*MT) == 0 and K % KC == 0 (holds: M=50000, MT=5).

template <int K, int NC, int MT>
__global__ __launch_bounds__((NC / 16) * 32)
void gemm_wmma_kernel(const float* __restrict__ A,
                      const float* __restrict__ W,
                      float* __restrict__ Out, int M) {
    constexpr int NW   = NC / 16;
    constexpr int TPB  = NW * 32;
    constexpr int ROWS = 16 * MT;
    constexpr int KC   = 128;               // K-chunk staged per buffer
    constexpr int LDKC = KC + 4;            // 16B-aligned rows, banks spread
    constexpr int NCH  = K / KC;
    constexpr int NBUF = (NCH > 1) ? 2 : 1;
    __shared__ float As[NBUF][ROWS * LDKC];

    const int m0 = blockIdx.x * ROWS;

    // Stage chunk `c` of the A strip into LDS buffer `buf`.
    auto stage = [&](int buf, int c) {
        const float* Ac = A + (size_t)m0 * K + c * KC;
        for (int i = threadIdx.x; i < ROWS * (KC / 4); i += TPB) {
            const int r  = i / (KC / 4);
            const int c4 = i % (KC / 4);
#if HAVE_ASYNC_LDS
            __builtin_amdgcn_global_load_async_to_lds_b128(
                (v4i_g*)(Ac + (size_t)r * K + c4 * 4),
                (v4i_l*)&As[buf][r * LDKC + c4 * 4],
                0, 0);
#else
            const float4 v = *(const float4*)(Ac + (size_t)r * K + c4 * 4);
            float* dp = &As[buf][r * LDKC + c4 * 4];
            dp[0] = v.x; dp[1] = v.y; dp[2] = v.z; dp[3] = v.w;
#endif
        }
    };

    const int lane = threadIdx.x & 31;
    const int wave = threadIdx.x >> 5;
    const int n0   = wave * 16;
    const int mm   = lane & 15;             // A row / B col within tile
    const int kk   = (lane >> 4) << 1;      // 0 (lanes 0-15) or 2 (16-31)
    const float* wrow = W + (size_t)(n0 + mm) * K;   // W[n0+mm, :]

    stage(0, 0);
    async_wait0();
    __syncthreads();

    v8f acc[MT] = {};
    for (int c = 0; c < NCH; ++c) {
        if (c + 1 < NCH) stage((c + 1) & (NBUF - 1), c + 1);   // overlap
        const float* as = As[c & (NBUF - 1)];
        const int kb = c * KC;
#pragma unroll 2
        for (int k4 = 0; k4 < KC; k4 += 4) {
            v2f b;
            b.x = wrow[kb + k4 + kk];
            b.y = wrow[kb + k4 + kk + 1];
#pragma unroll
            for (int t = 0; t < MT; ++t) {
                v2f a;
                const int base = (t * 16 + mm) * LDKC + k4 + kk;
                a.x = as[base];
                a.y = as[base + 1];
                acc[t] = __builtin_amdgcn_wmma_f32_16x16x4_f32(
                    /*neg_a=*/false, a, /*neg_b=*/false, b,
                    /*c_mod=*/(short)0, acc[t],
                    /*reuse_a=*/false, /*reuse_b=*/false);
            }
        }
        if (c + 1 < NCH) {
            async_wait0();
            __syncthreads();
        }
    }

    // C/D layout: VGPR r -> row (r + 8*(lane>>4)), col = lane&15
    const int rb = (lane >> 4) << 3;
#pragma unroll
    for (int t = 0; t < MT; ++t) {
        float* oc = Out + (size_t)(m0 + t * 16 + rb) * NC + n0 + mm;
#pragma unroll
        for (int r = 0; r < 8; ++r) oc[(size_t)r * NC] = acc[t][r];
    }
}

// ---------------------------------------------------------------------------

static inline int ceil_div(long long a, long long b) { return (int)((a + b - 1) / b); }

extern "C" void kernel_launch(void* const* d_in, const int* in_sizes, int n_in,
                              void* d_out, int out_size, void* d_ws, size_t ws_size,
                              hipStream_t stream) {
    const float* x    = (const float*)d_in[0];   // [M, 512]
    const int*   ei   = (const int*)d_in[1];     // [2, E]
    const float* W1   = (const float*)d_in[2];   // [128, 512]
    const float* b1   = (const float*)d_in[3];   // [128]
    const float* Wmu  = (const float*)d_in[4];   // [64, 128]
    const float* bmu  = (const float*)d_in[5];   // [64]
    const float* Wls  = (const float*)d_in[6];   // [64, 128]
    const float* bls  = (const float*)d_in[7];   // [64]
    float* out = (float*)d_out;

    const int M = in_sizes[0] / 512;             // 50000 (= 80 * 625)
    const int E = in_sizes[1] / 2;               // 800000
    const int* src = ei;
    const int* dst = ei + E;

    // Workspace layout (~54 MB):
    //   dis  : M floats     @ 0
    //   h1   : M*128 floats @ 1 MB   (reused as hm/hl in layer 2)
    //   agg1 : M*128 floats @ 28 MB  (becomes h2 after finalize)
    char* ws = (char*)d_ws;
    float* dis  = (float*)ws;
    float* h1   = (float*)(ws + ((size_t)1  << 20));
    float* agg1 = (float*)(ws + ((size_t)28 << 20));
    float* hm   = h1;                                   // layer-2 mu pre-agg
    float* hl   = (float*)(ws + ((size_t)14 << 20));    // layer-2 logstd pre-agg
    float* outmu = out;
    float* outls = out + (size_t)M * 64;

    // ---- degree + d^{-1/2} (self-loop counted via init to 1.0) ----
    fill_f32_kernel<<<ceil_div(M, 256), 256, 0, stream>>>(dis, M, 1.0f);
    deg_edges_kernel<<<ceil_div(E, 256), 256, 0, stream>>>(dst, dis, E);
    rsqrt_kernel<<<ceil_div(M, 256), 256, 0, stream>>>(dis, M);

    // ---- layer 1: h1 = x @ W1^T (WMMA fp32, 80 rows / block) ----
    gemm_wmma_kernel<512, 128, 5><<<M / 80, 256, 0, stream>>>(x, W1, h1, M);

    // ---- layer 1 aggregation: agg1 = scatter(h1*norm) + self + b1, ReLU ----
    fill_f32_kernel<<<ceil_div((long long)M * 128, 256), 256, 0, stream>>>(
        agg1, (long long)M * 128, 0.0f);
    scatter_kernel<128><<<ceil_div(E, 2), 256, 0, stream>>>(src, dst, dis, h1, agg1, E);
    finalize_kernel<<<ceil_div((long long)M * 128, 256), 256, 0, stream>>>(
        agg1, h1, dis, b1, (long long)M * 128, 128, /*relu=*/1);

    // ---- layer 2 GEMMs (h1's workspace is free now) ----
    gemm_wmma_kernel<128, 64, 5><<<M / 80, 128, 0, stream>>>(agg1, Wmu, hm, M);
    gemm_wmma_kernel<128, 64, 5><<<M / 80, 128, 0, stream>>>(agg1, Wls, hl, M);

    // ---- layer 2 aggregation into d_out ----
    fill_f32_kernel<<<ceil_div((long long)M * 128, 256), 256, 0, stream>>>(
        out, (long long)M * 128, 0.0f);   // mu + logstd = 2 * M*64
    scatter_kernel<64><<<ceil_div(E, 4), 256, 0, stream>>>(src, dst, dis, hm, outmu, E);
    scatter_kernel<64><<<ceil_div(E, 4), 256, 0, stream>>>(src, dst, dis, hl, outls, E);
    finalize_kernel<<<ceil_div((long long)M * 64, 256), 256, 0, stream>>>(
        outmu, hm, dis, bmu, (long long)M * 64, 64, /*relu=*/0);
    finalize_kernel<<<ceil_div((long long)M * 64, 256), 256, 0, stream>>>(
        outls, hl, dis, bls, (long long)M * 64, 64, /*relu=*/0);
}